// EmbedAtom_71992241815595
// MI455X (gfx1250) — compile-verified
//
#include <hip/hip_runtime.h>
#include <hip/hip_bf16.h>
#include <stdint.h>

// out[i, :] = W[:, atom_type[i]] + b   -> pure gather, HBM-write-bound (~512 MB out).
// Strategy:
//   k1: build fused transposed table  T[t*128 + e] = W[e*119 + t] + b[e]   (61 KB in d_ws)
//   k2: stage T into LDS via CDNA5 async global->LDS B128 loads (ASYNCcnt path), then
//       each wave32 emits one 512 B output row per step via ds_load_b128 +
//       non-temporal global_store_b128 (512 MB write-once stream >> 192 MB L2).

#define EMBED   128
#define NTYPES  119
#define TBL_F4  ((NTYPES * EMBED) / 4)   // 3808 float4 chunks

typedef __attribute__((ext_vector_type(4))) float v4f;   // native vector: OK for nontemporal builtins

__global__ void build_table_kernel(const float* __restrict__ W,
                                   const float* __restrict__ b,
                                   float* __restrict__ T) {
    int i = blockIdx.x * blockDim.x + threadIdx.x;
    if (i < NTYPES * EMBED) {
        int t = i >> 7;        // i / 128
        int e = i & 127;       // i % 128
        T[i] = W[e * NTYPES + t] + b[e];
    }
}

__global__ void gather_rows_kernel(const int* __restrict__ atype,
                                   const float* __restrict__ T,
                                   float* __restrict__ out,
                                   int n) {
    __shared__ float tile[NTYPES * EMBED];   // 60928 bytes: < 64 KB static limit, << 320 KB/WGP

    const int tid = threadIdx.x;

    // ---- Stage table into LDS with async global->LDS B128 copies (ASYNCcnt path) ----
    {
        // Low 32 bits of a generic pointer into the LDS aperture == LDS byte offset.
        unsigned lds_base = (unsigned)(unsigned long long)(uintptr_t)(&tile[0]);
        unsigned long long gbase = (unsigned long long)(uintptr_t)T;
        for (int i = tid; i < TBL_F4; i += blockDim.x) {
            unsigned lds_off = lds_base + (unsigned)i * 16u;
            unsigned long long gaddr = gbase + (unsigned long long)i * 16ull;
            asm volatile("global_load_async_to_lds_b128 %0, %1, off"
                         :: "v"(lds_off), "v"(gaddr)
                         : "memory");
        }
        asm volatile("s_wait_asynccnt 0" ::: "memory");
        __syncthreads();
    }

    const int lane   = tid & 31;
    const int wave   = (blockIdx.x * blockDim.x + tid) >> 5;
    const int nwaves = (gridDim.x * blockDim.x) >> 5;

    const v4f* __restrict__ tile4 = reinterpret_cast<const v4f*>(tile);
    v4f* __restrict__ out4        = reinterpret_cast<v4f*>(out);

    // Each wave handles 32 atoms per outer step: lanes fetch 32 ids at once
    // (coalesced, non-temporal: read-once stream), then broadcast each id;
    // lane l moves float4 chunk l of the selected row.
    for (int base = wave * 32; base < n; base += nwaves * 32) {
        int myAtom = base + lane;
        int myT    = (myAtom < n) ? __builtin_nontemporal_load(atype + myAtom) : 0;
        int limit  = n - base;
        if (limit > 32) limit = 32;
        for (int j = 0; j < limit; ++j) {
            int t = __shfl(myT, j, 32);                 // wave-uniform row id
            v4f v = tile4[t * 32 + lane];               // ds_load_b128, bank-conflict-free
            // 512 MB write-once stream: non-temporal so L2 keeps reused data
            __builtin_nontemporal_store(v, &out4[(size_t)(base + j) * 32 + lane]);
        }
    }
}

extern "C" void kernel_launch(void* const* d_in, const int* in_sizes, int n_in,
                              void* d_out, int out_size, void* d_ws, size_t ws_size,
                              hipStream_t stream) {
    const int*   atype = (const int*)d_in[0];
    const float* W     = (const float*)d_in[1];
    const float* b     = (const float*)d_in[2];
    float*       out   = (float*)d_out;
    float*       table = (float*)d_ws;          // 61 KB scratch table
    const int n = in_sizes[0];

    build_table_kernel<<<(NTYPES * EMBED + 255) / 256, 256, 0, stream>>>(W, b, table);
    gather_rows_kernel<<<2048, 256, 0, stream>>>(atype, table, out, n);
}